// Encoder_72335839199981
// MI455X (gfx1250) — compile-verified
//
#include <hip/hip_runtime.h>
#include <math.h>

typedef __attribute__((ext_vector_type(2))) float v2f;
typedef __attribute__((ext_vector_type(8))) float v8f;

#define LDA1 258   // 16 x 256 A-tile, padded (even stride: 8B-aligned K pairs, conflict-free)
#define LDA2 130   // 16 x 128 A-tile, padded

// ---------------- GEMM1: h = x @ W1  (N x 256) * (256 x 128) ----------------
__global__ __launch_bounds__(256) void gemm1_kernel(const float* __restrict__ x,
                                                    const float* __restrict__ W1,
                                                    float* __restrict__ h, int N_) {
    __shared__ float lds[16 * LDA1];
    const int tile = blockIdx.x;
    const int t    = threadIdx.x;
    const int row0 = tile * 16;
    const bool full = (row0 + 16 <= N_);       // block-uniform

    // Stage 16x256 x-tile (coalesced: thread t loads column t of each row)
    #pragma unroll
    for (int i = 0; i < 16; ++i) {
        int r = row0 + i;
        if (r >= N_) r = N_ - 1;               // clamp (pad rows harmless)
        lds[i * LDA1 + t] = x[(size_t)r * 256 + t];
    }
    __syncthreads();

    const int wave = t >> 5;                   // 0..7 -> column tile
    const int lane = t & 31;
    const int m    = lane & 15;                // A row within tile
    const int koff = (lane >> 4) * 2;          // K sub-offset per lane half
    const int n    = (wave << 4) + m;          // output column (0..127)

    v8f acc = {};
    for (int k0 = 0; k0 < 256; k0 += 4) {
        v2f a, b;
        a.x = lds[m * LDA1 + k0 + koff];
        a.y = lds[m * LDA1 + k0 + koff + 1];
        b.x = W1[(size_t)(k0 + koff)     * 128 + n];
        b.y = W1[(size_t)(k0 + koff + 1) * 128 + n];
        acc = __builtin_amdgcn_wmma_f32_16x16x4_f32(false, a, false, b,
                                                    (short)0, acc, false, false);
    }

    // D layout: VGPR r -> row (r + 8*(lane>=16)), col n
    const int rbase = row0 + ((lane >> 4) << 3);
    if (full) {                                 // single scalar branch, no exec juggling
        #pragma unroll
        for (int r = 0; r < 8; ++r)
            h[(size_t)(rbase + r) * 128 + n] = acc[r];
    } else {
        #pragma unroll
        for (int r = 0; r < 8; ++r) {
            const int row = rbase + r;
            if (row < N_) h[(size_t)row * 128 + n] = acc[r];
        }
    }
}

// ---------------- attention logits: e_src/e_dst = h . att ----------------
__global__ __launch_bounds__(256) void attn_kernel(const float* __restrict__ h,
                                                   const float* __restrict__ att_src,
                                                   const float* __restrict__ att_dst,
                                                   float* __restrict__ e_src,
                                                   float* __restrict__ e_dst, int N_) {
    const int node = (int)((blockIdx.x * (unsigned)blockDim.x + threadIdx.x) >> 5);
    const int lane = threadIdx.x & 31;
    if (node >= N_) return;
    const float4 hv = ((const float4*)(h + (size_t)node * 128))[lane];
    const float4 as = ((const float4*)att_src)[lane];
    const float4 ad = ((const float4*)att_dst)[lane];
    float s = hv.x * as.x + hv.y * as.y + hv.z * as.z + hv.w * as.w;
    float d = hv.x * ad.x + hv.y * ad.y + hv.z * ad.z + hv.w * ad.w;
    #pragma unroll
    for (int o = 16; o > 0; o >>= 1) {
        s += __shfl_down(s, o, 32);
        d += __shfl_down(d, o, 32);
    }
    if (lane == 0) { e_src[node] = s; e_dst[node] = d; }
}

// ---------------- per-edge alpha + segment max (bitwise atomicMax, alpha>0) ----------------
__global__ __launch_bounds__(256) void edge_alpha_max(const int* __restrict__ ei,
                                                      const float* __restrict__ e_src,
                                                      const float* __restrict__ e_dst,
                                                      float* __restrict__ alpha,
                                                      unsigned* __restrict__ mbits, int E_) {
    const int e = blockIdx.x * blockDim.x + threadIdx.x;
    if (e >= E_) return;
    const int s = ei[e];
    const int d = ei[E_ + e];
    float a = e_src[s] + e_dst[d];
    a = 1.0f / (1.0f + expf(-a));              // sigmoid, in (0,1)
    alpha[e] = a;
    atomicMax(mbits + d, __float_as_uint(a));  // positive floats: int order == float order
}

// ---------------- exp(alpha - m[dst]) and segment sum ----------------
__global__ __launch_bounds__(256) void edge_expsum(const int* __restrict__ ei,
                                                   const float* __restrict__ alpha,
                                                   const unsigned* __restrict__ mbits,
                                                   float* __restrict__ denom,
                                                   float* __restrict__ exb, int E_) {
    const int e = blockIdx.x * blockDim.x + threadIdx.x;
    if (e >= E_) return;
    const int d = ei[E_ + e];
    const float mv = __uint_as_float(mbits[d]);
    const float ex = expf(alpha[e] - mv);
    exb[e] = ex;
    atomicAdd(denom + d, ex);
}

// ---------------- weighted scatter-add aggregation (wave per edge) ----------------
__global__ __launch_bounds__(256) void edge_agg(const int* __restrict__ ei,
                                                const float* __restrict__ h,
                                                const float* __restrict__ exb,
                                                const float* __restrict__ denom,
                                                float* __restrict__ out1, int E_) {
    const int e    = (int)((blockIdx.x * (unsigned)blockDim.x + threadIdx.x) >> 5);
    const int lane = threadIdx.x & 31;
    if (e >= E_) return;
    const int s = ei[e];
    const int d = ei[E_ + e];
    const float w = exb[e] / denom[d];
    const float4 hv = ((const float4*)(h + (size_t)s * 128))[lane];
    float* o = out1 + (size_t)d * 128 + lane * 4;
    atomicAdd(o + 0, hv.x * w);
    atomicAdd(o + 1, hv.y * w);
    atomicAdd(o + 2, hv.z * w);
    atomicAdd(o + 3, hv.w * w);
}

// ---------------- GEMM2: out = elu(out1) @ W2  (N x 128) * (128 x 128) ----------------
__global__ __launch_bounds__(256) void gemm2_kernel(const float* __restrict__ out1,
                                                    const float* __restrict__ W2,
                                                    float* __restrict__ out, int N_) {
    __shared__ float lds[16 * LDA2];
    const int tile = blockIdx.x;
    const int t    = threadIdx.x;
    const int row0 = tile * 16;
    const bool full = (row0 + 16 <= N_);       // block-uniform

    // Stage 16x128 elu(out1) tile: 2048 elems / 256 threads = 8 each
    #pragma unroll
    for (int i = 0; i < 8; ++i) {
        const int idx = i * 256 + t;
        const int r   = idx >> 7;
        const int c   = idx & 127;
        int row = row0 + r;
        if (row >= N_) row = N_ - 1;
        float v = out1[(size_t)row * 128 + c];
        lds[r * LDA2 + c] = (v > 0.0f) ? v : (expf(v) - 1.0f);   // ELU
    }
    __syncthreads();

    const int wave = t >> 5;
    const int lane = t & 31;
    const int m    = lane & 15;
    const int koff = (lane >> 4) * 2;
    const int n    = (wave << 4) + m;

    v8f acc = {};
    for (int k0 = 0; k0 < 128; k0 += 4) {
        v2f a, b;
        a.x = lds[m * LDA2 + k0 + koff];
        a.y = lds[m * LDA2 + k0 + koff + 1];
        b.x = W2[(size_t)(k0 + koff)     * 128 + n];
        b.y = W2[(size_t)(k0 + koff + 1) * 128 + n];
        acc = __builtin_amdgcn_wmma_f32_16x16x4_f32(false, a, false, b,
                                                    (short)0, acc, false, false);
    }

    const int rbase = row0 + ((lane >> 4) << 3);
    if (full) {
        #pragma unroll
        for (int r = 0; r < 8; ++r)
            out[(size_t)(rbase + r) * 128 + n] = acc[r];
    } else {
        #pragma unroll
        for (int r = 0; r < 8; ++r) {
            const int row = rbase + r;
            if (row < N_) out[(size_t)row * 128 + n] = acc[r];
        }
    }
}

extern "C" void kernel_launch(void* const* d_in, const int* in_sizes, int n_in,
                              void* d_out, int out_size, void* d_ws, size_t ws_size,
                              hipStream_t stream) {
    const float* x       = (const float*)d_in[0];
    const int*   ei      = (const int*)  d_in[1];
    const float* W1      = (const float*)d_in[2];
    const float* att_src = (const float*)d_in[3];
    const float* att_dst = (const float*)d_in[4];
    const float* W2      = (const float*)d_in[5];
    float*       out     = (float*)d_out;

    const int N_ = in_sizes[0] / 256;   // 50000
    const int E_ = in_sizes[1] / 2;     // 800000

    // workspace layout (floats)
    float*    ws    = (float*)d_ws;
    float*    h     = ws;                                  // N*128
    float*    out1  = h     + (size_t)N_ * 128;            // N*128
    float*    e_src = out1  + (size_t)N_ * 128;            // N
    float*    e_dst = e_src + N_;                          // N
    unsigned* mbits = (unsigned*)(e_dst + N_);             // N
    float*    denom = (float*)(mbits + N_);                // N
    float*    alpha = denom + N_;                          // E
    float*    exb   = alpha + E_;                          // E

    hipMemsetAsync(out1,  0, (size_t)N_ * 128 * sizeof(float), stream);
    hipMemsetAsync(mbits, 0, (size_t)N_ * sizeof(unsigned),    stream);
    hipMemsetAsync(denom, 0, (size_t)N_ * sizeof(float),       stream);

    const int tiles = (N_ + 15) / 16;
    gemm1_kernel  <<<tiles, 256, 0, stream>>>(x, W1, h, N_);
    attn_kernel   <<<(N_ * 32 + 255) / 256, 256, 0, stream>>>(h, att_src, att_dst,
                                                              e_src, e_dst, N_);
    edge_alpha_max<<<(E_ + 255) / 256, 256, 0, stream>>>(ei, e_src, e_dst, alpha, mbits, E_);
    edge_expsum   <<<(E_ + 255) / 256, 256, 0, stream>>>(ei, alpha, mbits, denom, exb, E_);
    edge_agg      <<<(E_ * 32 + 255) / 256, 256, 0, stream>>>(ei, h, exb, denom, out1, E_);
    gemm2_kernel  <<<tiles, 256, 0, stream>>>(out1, W2, out, N_);
}